// decoder2DStage2_70205535420638
// MI455X (gfx1250) — compile-verified
//
#include <hip/hip_runtime.h>

// fp32 WMMA fragment types (16x16x4: A=16x4 -> 2 VGPRs, C/D=16x16 -> 8 VGPRs)
typedef float v2f __attribute__((ext_vector_type(2)));
typedef float v8f __attribute__((ext_vector_type(8)));

#define ROWS_PER_BLOCK 32          // 2 waves x 16 rows
#define ROW_STRIDE     260         // 256 + 4 pad floats: conflict-free ds_load_b64, rows stay 16B aligned

__global__ __launch_bounds__(64) void decoder2d_min_mask_kernel(
    const float* __restrict__ in,     // [rows, 256]
    const float* __restrict__ Tg,     // [256, 64]
    float* __restrict__ out_x3,       // [rows]
    float* __restrict__ out_M)        // [rows]
{
    __shared__ float lds_in[ROWS_PER_BLOCK * ROW_STRIDE];

    const int tid  = threadIdx.x;          // 0..63
    const int wave = tid >> 5;             // 0..1
    const int lane = tid & 31;
    const int half = lane >> 4;            // K-half selector for A/B fragments
    const int m    = lane & 15;            // M index (A) / N index (B)

    const long row0 = (long)blockIdx.x * ROWS_PER_BLOCK;
    const float* gsrc = in + row0 * 256;

    // ---- Stage 32 rows x 256 fp32 (contiguous 32 KB) into LDS with relu ----
    #pragma unroll
    for (int it = 0; it < 32; ++it) {
        const int f = (it * 64 + tid) * 4;             // flat float index, 16B aligned
        const float4 v = *(const float4*)(gsrc + f);
        const int r = f >> 8;
        const int k = f & 255;
        float* dst = &lds_in[r * ROW_STRIDE + k];
        dst[0] = fmaxf(v.x, 0.f);
        dst[1] = fmaxf(v.y, 0.f);
        dst[2] = fmaxf(v.z, 0.f);
        dst[3] = fmaxf(v.w, 0.f);
    }
    __syncthreads();

    // ---- GEMM tile: D(16x64) = A(16x256) * T(256x64), fp32 WMMA 16x16x4 ----
    // A fragment (per K-step kk): lanes 0-15 hold (M=m, K=4kk+{0,1}),
    //                             lanes 16-31 hold (M=m, K=4kk+{2,3})
    const float* aBase = &lds_in[(wave * 16 + m) * ROW_STRIDE + 2 * half];
    // B fragment: lanes 0-15 hold (K=4kk+{0,1}, N), lanes 16-31 (K=4kk+{2,3}, N)
    //   T index = (4kk + 2*half + v)*64 + 16n + m  ->  base(128*half + m) + 1024B*kk + 256B*v + 64B*n
    const float* bBase = Tg + half * 128 + m;

    v8f acc0 = {}, acc1 = {}, acc2 = {}, acc3 = {};

    #pragma unroll 8
    for (int kk = 0; kk < 64; ++kk) {
        const v2f a = *(const v2f*)(aBase + 4 * kk);   // ds_load_b64, conflict-free
        const float* bk = bBase + 256 * kk;            // L2/WGP$-resident (T is 64 KB)
        v2f b0, b1, b2, b3;
        b0.x = bk[0];   b0.y = bk[64];
        b1.x = bk[16];  b1.y = bk[80];
        b2.x = bk[32];  b2.y = bk[96];
        b3.x = bk[48];  b3.y = bk[112];
        // 4 independent accumulation chains -> deep WMMA pipelining
        acc0 = __builtin_amdgcn_wmma_f32_16x16x4_f32(false, a, false, b0, (short)0, acc0, false, false);
        acc1 = __builtin_amdgcn_wmma_f32_16x16x4_f32(false, a, false, b1, (short)0, acc1, false, false);
        acc2 = __builtin_amdgcn_wmma_f32_16x16x4_f32(false, a, false, b2, (short)0, acc2, false, false);
        acc3 = __builtin_amdgcn_wmma_f32_16x16x4_f32(false, a, false, b3, (short)0, acc3, false, false);
    }

    // ---- Per-lane combine across the 4 N-tiles ----
    // C layout: VGPR i = (M = i + 8*half, N = m). So component i + lane-half pick a row,
    // and the 16 lanes of each half span 16 N values of that row.
    float rmin[8];
    int   rcnt[8];
    #pragma unroll
    for (int i = 0; i < 8; ++i) {
        const float x0 = acc0[i], x1 = acc1[i], x2 = acc2[i], x3v = acc3[i];
        rmin[i] = fminf(fminf(x0, x1), fminf(x2, x3v));
        rcnt[i] = (x0 != 0.f) + (x1 != 0.f) + (x2 != 0.f) + (x3v != 0.f);
    }

    // ---- Cross-lane reduction within each 16-lane half (covers all 64 N) ----
    #pragma unroll
    for (int i = 0; i < 8; ++i) {
        #pragma unroll
        for (int d = 1; d <= 8; d <<= 1) {
            rmin[i] = fminf(rmin[i], __shfl_xor(rmin[i], d, 32));
            rcnt[i] += __shfl_xor(rcnt[i], d, 32);
        }
    }

    // Lanes 0 and 16 hold results for rows (wave*16 + half*8 + i)
    if (m == 0) {
        const long base = row0 + wave * 16 + half * 8;
        #pragma unroll
        for (int i = 0; i < 8; ++i) {
            out_x3[base + i] = rmin[i];
            // M = clip(64 - nnz - 1, 0, 1): nnz in {63,64} -> 0, else 1
            out_M[base + i] = (rcnt[i] >= 63) ? 0.f : 1.f;
        }
    }
}

extern "C" void kernel_launch(void* const* d_in, const int* in_sizes, int n_in,
                              void* d_out, int out_size, void* d_ws, size_t ws_size,
                              hipStream_t stream) {
    const float* in = (const float*)d_in[0];   // [32, 16384, 256] fp32
    const float* Tg = (const float*)d_in[1];   // [256, 64] fp32 (binary)
    const int total_rows = in_sizes[0] / 256;  // 524288
    float* x3 = (float*)d_out;                 // first output, [rows]
    float* M  = x3 + total_rows;               // second output, [rows]

    const int blocks = total_rows / ROWS_PER_BLOCK;  // 16384
    decoder2d_min_mask_kernel<<<blocks, 64, 0, stream>>>(in, Tg, x3, M);
}